// SpookyLocalBlock_52879637348406
// MI455X (gfx1250) — compile-verified
//
#include <hip/hip_runtime.h>
#include <math.h>

typedef __attribute__((ext_vector_type(2))) float v2f;
typedef __attribute__((ext_vector_type(8))) float v8f;

#define NATOM 20000
#define FDIM  128
#define NJ    12
#define KDIM  16
#define BN_BLOCKS 100
#define LDST  132   // padded LDS stride for 128-wide rows (bank-conflict free, 16B aligned)
#define GSTRIDE 17  // padded stride for G [128 x 16]
#define ASTRIDE 20  // padded stride for per-atom A matrices [144 x 12] (8B aligned, conflict free)

// workspace layout (float offsets); total ~10.3M floats (~41.2 MB)
#define WS_PART  0
#define WS_SCALE (WS_PART + 2*BN_BLOCKS*FDIM)
#define WS_SHIFT (WS_SCALE + 4*FDIM)
#define WS_QP    (WS_SHIFT + 4*FDIM)
#define WS_QD    (WS_QP + FDIM*FDIM)
#define WS_FEAT  (WS_QD + FDIM*FDIM)   // 4 tables of [N,128]: c,s,p,d

__device__ __forceinline__ float silu(float x) { return x / (1.0f + __expf(-x)); }

// D = A(16x4 f32) * B(4x16 f32) + C(16x16 f32)
__device__ __forceinline__ v8f wmma_f32x4(v2f a, v2f b, v8f c) {
  return __builtin_amdgcn_wmma_f32_16x16x4_f32(false, a, false, b, (short)0, c, false, false);
}

// ---------------- Kernel 1: BN partial sums (deterministic two-pass) ----------------
__global__ __launch_bounds__(256) void bn_partial(const float* __restrict__ x,
                                                  float* __restrict__ ws) {
  __shared__ float p1[256], p2[256];
  const int tid = threadIdx.x;
  const int f = tid & 127, rh = tid >> 7;
  const int r0 = blockIdx.x * 200;
  const int r1 = (r0 + 200 < NATOM) ? r0 + 200 : NATOM;
  float s = 0.f, q = 0.f;
  for (int r = r0 + rh; r < r1; r += 2) {
    float v = x[(size_t)r * FDIM + f];
    s += v; q += v * v;
  }
  p1[tid] = s; p2[tid] = q;
  __syncthreads();
  if (tid < 128) {
    ws[WS_PART + blockIdx.x * FDIM + f]                    = p1[tid] + p1[tid + 128];
    ws[WS_PART + BN_BLOCKS * FDIM + blockIdx.x * FDIM + f] = p2[tid] + p2[tid + 128];
  }
}

// ---------------- Kernel 2: BN finalize -> per-mlp scale/shift ----------------
__global__ __launch_bounds__(128) void bn_finalize(const float* __restrict__ bng,
                                                   const float* __restrict__ bnb,
                                                   float* __restrict__ ws) {
  const int f = threadIdx.x;
  float s = 0.f, q = 0.f;
  for (int b = 0; b < BN_BLOCKS; b++) {
    s += ws[WS_PART + b * FDIM + f];
    q += ws[WS_PART + BN_BLOCKS * FDIM + b * FDIM + f];
  }
  const float inv = 1.0f / (float)NATOM;
  const float mu  = s * inv;
  const float var = q * inv - mu * mu;
  const float ri  = rsqrtf(var + 1e-5f);
  for (int m = 0; m < 4; m++) {
    float sc = bng[m * FDIM + f] * ri;
    ws[WS_SCALE + m * FDIM + f] = sc;
    ws[WS_SHIFT + m * FDIM + f] = bnb[m * FDIM + f] - sc * mu;
  }
}

// ---------------- Kernel 3: Q = M2 @ M1^T  (Qp, Qd) ----------------
__global__ __launch_bounds__(128) void qmat(const float* __restrict__ P1, const float* __restrict__ P2,
                                            const float* __restrict__ D1, const float* __restrict__ D2,
                                            float* __restrict__ ws) {
  const int b = blockIdx.x;
  const int which = b >> 7;
  const int g = b & 127;
  const float* M2 = which ? D2 : P2;
  const float* M1 = which ? D1 : P1;
  float* Q = ws + (which ? WS_QD : WS_QP);
  const int h = threadIdx.x;
  float acc = 0.f;
  for (int f = 0; f < FDIM; f++) acc += M2[g * FDIM + f] * M1[h * FDIM + f];
  Q[g * FDIM + h] = acc;
}

// ---------------- Kernel 4: fused ResMLP via f32 WMMA ----------------
// out = silu( bn(x) + silu(bn(x)) @ W1^T + b1 ) @ W2^T + b2   -> feature table m
__global__ __launch_bounds__(256) void resmlp_wmma(const float* __restrict__ x,
                                                   const float* __restrict__ resW,
                                                   const float* __restrict__ resb,
                                                   const float* __restrict__ linW,
                                                   const float* __restrict__ linb,
                                                   float* __restrict__ ws) {
  __shared__ __align__(16) float abuf[128 * LDST];  // A operand (silu activations)
  __shared__ __align__(16) float wbuf[128 * LDST];  // weight matrix, row-major [f][g]
  const int tid  = threadIdx.x;
  const int m    = blockIdx.y;
  const int row0 = blockIdx.x * 128;
  const float* scale = ws + WS_SCALE + m * FDIM;
  const float* shift = ws + WS_SHIFT + m * FDIM;
  const float* rb = resb + m * FDIM;
  const float* lb = linb + m * FDIM;
  const float* W1 = resW + (size_t)m * FDIM * FDIM;
  const float* W2 = linW + (size_t)m * FDIM * FDIM;
  float* feat = ws + WS_FEAT + (size_t)m * NATOM * FDIM;

  // stage silu(bn(x)) and W1 (128x128 each), float4 coalesced
  for (int i = 0; i < 16; i++) {
    int id4 = i * 256 + tid;
    int r = id4 >> 5, c4 = (id4 & 31) * 4;
    float4 v = make_float4(0.f, 0.f, 0.f, 0.f);
    if (row0 + r < NATOM) v = *(const float4*)(x + (size_t)(row0 + r) * FDIM + c4);
    float4 sv;
    sv.x = silu(v.x * scale[c4 + 0] + shift[c4 + 0]);
    sv.y = silu(v.y * scale[c4 + 1] + shift[c4 + 1]);
    sv.z = silu(v.z * scale[c4 + 2] + shift[c4 + 2]);
    sv.w = silu(v.w * scale[c4 + 3] + shift[c4 + 3]);
    *(float4*)(abuf + r * LDST + c4) = sv;
    *(float4*)(wbuf + r * LDST + c4) = *(const float4*)(W1 + r * FDIM + c4);
  }
  __syncthreads();

  const int lane = tid & 31;
  const int wave = tid >> 5;
  const int ln = lane & 15, h = lane >> 4;
  const int mrow = wave * 16;          // this wave's 16-row strip
  v8f acc[8];

  // GEMM1: C init = xh + res_b (residual folded into accumulator)
  for (int nt = 0; nt < 8; nt++) {
    int n = nt * 16 + ln;
    float sc = scale[n], sh = shift[n], rn = rb[n];
    v8f c;
    for (int r = 0; r < 8; r++) {
      int row = row0 + mrow + r + 8 * h;
      float xv = (row < NATOM) ? x[(size_t)row * FDIM + n] : 0.f;
      c[r] = xv * sc + sh + rn;
    }
    acc[nt] = c;
  }
  for (int ks = 0; ks < 32; ks++) {
    int kb = ks * 4 + 2 * h;
    v2f a = *(const v2f*)(abuf + (mrow + ln) * LDST + kb);
#pragma unroll
    for (int nt = 0; nt < 8; nt++) {
      v2f b = *(const v2f*)(wbuf + (nt * 16 + ln) * LDST + kb);  // B[k][n] = W1[n][k]
      acc[nt] = wmma_f32x4(a, b, acc[nt]);
    }
  }
  __syncthreads();

  // epilogue1: abuf <- silu(r); wbuf <- W2
  for (int nt = 0; nt < 8; nt++)
    for (int r = 0; r < 8; r++)
      abuf[(mrow + r + 8 * h) * LDST + nt * 16 + ln] = silu(acc[nt][r]);
  for (int i = 0; i < 16; i++) {
    int id4 = i * 256 + tid;
    int r = id4 >> 5, c4 = (id4 & 31) * 4;
    *(float4*)(wbuf + r * LDST + c4) = *(const float4*)(W2 + r * FDIM + c4);
  }
  __syncthreads();

  // GEMM2: C init = lin_b
  for (int nt = 0; nt < 8; nt++) {
    float bn = lb[nt * 16 + ln];
    v8f c;
    for (int r = 0; r < 8; r++) c[r] = bn;
    acc[nt] = c;
  }
  for (int ks = 0; ks < 32; ks++) {
    int kb = ks * 4 + 2 * h;
    v2f a = *(const v2f*)(abuf + (mrow + ln) * LDST + kb);
#pragma unroll
    for (int nt = 0; nt < 8; nt++) {
      v2f b = *(const v2f*)(wbuf + (nt * 16 + ln) * LDST + kb);
      acc[nt] = wmma_f32x4(a, b, acc[nt]);
    }
  }
  for (int nt = 0; nt < 8; nt++)
    for (int r = 0; r < 8; r++) {
      int row = row0 + mrow + r + 8 * h;
      if (row < NATOM) feat[(size_t)row * FDIM + nt * 16 + ln] = acc[nt][r];
    }
}

// ---------------- Kernel 5: env contraction (WMMA) + quadratic forms + output ----------------
// Per atom i:
//   m2[(k,c), f] = sum_j g[j,k,c] * feat_type[nbr_j, f]   (WMMA: [144,12]@[12,128])
//   all[c, f]    = sum_k G[f,k] * m2[(k,c), f]
//   final_p = sum_c  all_p[c] Qp all_p[c]^T ; final_d likewise
//   out[i,f] = c_fea[i,f] + all_s[f] + final_p + final_d
__global__ __launch_bounds__(256) void env_final(const int* __restrict__ nbr,
                                                 const float* __restrict__ gs,
                                                 const float* __restrict__ gp,
                                                 const float* __restrict__ gd,
                                                 const float* __restrict__ Gs,
                                                 const float* __restrict__ Gp,
                                                 const float* __restrict__ Gd,
                                                 const float* __restrict__ ws,
                                                 float* __restrict__ out) {
  __shared__ __align__(16) float gbuf[144 * ASTRIDE];      // A matrices (rows: 16 s | 48 p | 80 d)
  __shared__ __align__(16) float featb[3 * NJ * LDST];     // gathered neighbor features (B)
  __shared__ float m2buf[144 * FDIM];
  __shared__ float Gbuf[3 * FDIM * GSTRIDE];
  __shared__ float allb[9 * FDIM];                         // ch0=s, ch1-3=p, ch4-8=d
  __shared__ float tbuf[2 * 8 * FDIM];
  __shared__ float red[512];
  __shared__ int   sidx[NJ];

  const int i   = blockIdx.x;
  const int tid = threadIdx.x;

  if (tid < NJ) sidx[tid] = nbr[(size_t)i * NJ + tid];

  // stage G tables (padded)
  for (int el = tid; el < 3 * FDIM * KDIM; el += 256) {
    int t = el >> 11, rem = el & 2047;
    int f = rem >> 4, k = rem & 15;
    const float* G = (t == 0) ? Gs : ((t == 1) ? Gp : Gd);
    Gbuf[t * FDIM * GSTRIDE + f * GSTRIDE + k] = G[f * KDIM + k];
  }
  // stage per-atom A matrices: A[row][j]
  const float* gsi = gs + (size_t)i * NJ * KDIM;
  const float* gpi = gp + (size_t)i * NJ * KDIM * 3;
  const float* gdi = gd + (size_t)i * NJ * KDIM * 5;
  if (tid < NJ * KDIM) {                                    // s: row = k
    int j = tid >> 4, k = tid & 15;
    gbuf[k * ASTRIDE + j] = gsi[tid];
  }
  for (int el = tid; el < NJ * KDIM * 3; el += 256) {       // p: row = 16 + k*3 + c
    int j = el / 48, r = el % 48;
    gbuf[(16 + r) * ASTRIDE + j] = gpi[el];
  }
  for (int el = tid; el < NJ * KDIM * 5; el += 256) {       // d: row = 64 + k*5 + c
    int j = el / 80, r = el % 80;
    gbuf[(64 + r) * ASTRIDE + j] = gdi[el];
  }
  __syncthreads();

  // gather neighbor features from L2-resident tables (tables 1..3 = s,p,d)
  const float* feat = ws + WS_FEAT;
  for (int el = tid; el < 3 * NJ * FDIM; el += 256) {
    int t = el / (NJ * FDIM);
    int rem = el - t * NJ * FDIM;
    int j = rem >> 7, f = rem & 127;
    featb[t * NJ * LDST + j * LDST + f] =
        feat[(size_t)(t + 1) * NATOM * FDIM + (size_t)sidx[j] * FDIM + f];
  }
  __syncthreads();

  // WMMA: 9 M-tiles x 8 N-tiles x 3 k-steps (K=12 exact)
  {
    const int lane = tid & 31, wave = tid >> 5;
    const int ln = lane & 15, h = lane >> 4;
    for (int mt = wave; mt < 9; mt += 8) {
      int t = (mt == 0) ? 0 : ((mt < 4) ? 1 : 2);
      const float* fb = featb + t * NJ * LDST;
      v2f a0 = *(const v2f*)(gbuf + (mt * 16 + ln) * ASTRIDE + 0 + 2 * h);
      v2f a1 = *(const v2f*)(gbuf + (mt * 16 + ln) * ASTRIDE + 4 + 2 * h);
      v2f a2 = *(const v2f*)(gbuf + (mt * 16 + ln) * ASTRIDE + 8 + 2 * h);
      for (int nt = 0; nt < 8; nt++) {
        int n = nt * 16 + ln;
        v8f c = {0.f, 0.f, 0.f, 0.f, 0.f, 0.f, 0.f, 0.f};
        v2f b;
        b.x = fb[(0 + 2 * h) * LDST + n]; b.y = fb[(1 + 2 * h) * LDST + n];
        c = wmma_f32x4(a0, b, c);
        b.x = fb[(4 + 2 * h) * LDST + n]; b.y = fb[(5 + 2 * h) * LDST + n];
        c = wmma_f32x4(a1, b, c);
        b.x = fb[(8 + 2 * h) * LDST + n]; b.y = fb[(9 + 2 * h) * LDST + n];
        c = wmma_f32x4(a2, b, c);
        for (int r = 0; r < 8; r++)
          m2buf[(mt * 16 + r + 8 * h) * FDIM + n] = c[r];
      }
    }
  }
  __syncthreads();

  // all[ch, f] = sum_k G[f,k] * m2[row(k,ch), f]
  {
    int f = tid & 127, grp = tid >> 7;
    int c0 = grp ? 4 : 0, c1 = grp ? 9 : 4;
    for (int ch = c0; ch < c1; ch++) {
      int t   = (ch == 0) ? 0 : ((ch < 4) ? 1 : 2);
      int rb_ = (ch == 0) ? 0 : ((ch < 4) ? (16 + ch - 1) : (64 + ch - 4));
      int rs_ = (ch == 0) ? 1 : ((ch < 4) ? 3 : 5);
      const float* Gt = Gbuf + t * FDIM * GSTRIDE + f * GSTRIDE;
      float acc = 0.f;
#pragma unroll
      for (int k = 0; k < 16; k++) acc += Gt[k] * m2buf[(rb_ + k * rs_) * FDIM + f];
      allb[ch * FDIM + f] = acc;
    }
  }
  __syncthreads();

  // t[c][g] = sum_f all[c,f] * Q[f,g] (Qp for c<3, Qd for c>=3); split f-range over 2 halves
  const float* Qp = ws + WS_QP;
  const float* Qd = ws + WS_QD;
  {
    int g = tid & 127, half = tid >> 7;
    float t8[8] = {0.f, 0.f, 0.f, 0.f, 0.f, 0.f, 0.f, 0.f};
    for (int ff = 0; ff < 64; ff++) {
      int f = half * 64 + ff;
      float qp = Qp[f * FDIM + g];
      float qd = Qd[f * FDIM + g];
      t8[0] += allb[1 * FDIM + f] * qp;
      t8[1] += allb[2 * FDIM + f] * qp;
      t8[2] += allb[3 * FDIM + f] * qp;
      t8[3] += allb[4 * FDIM + f] * qd;
      t8[4] += allb[5 * FDIM + f] * qd;
      t8[5] += allb[6 * FDIM + f] * qd;
      t8[6] += allb[7 * FDIM + f] * qd;
      t8[7] += allb[8 * FDIM + f] * qd;
    }
    for (int c = 0; c < 8; c++) tbuf[half * 1024 + c * FDIM + g] = t8[c];
  }
  __syncthreads();

  // final dot: sum_g (t0+t1)[c][g] * all[c][g], split p vs d, deterministic tree reduce
  {
    float pP = 0.f, dP = 0.f;
    for (int ii = 0; ii < 4; ii++) {
      int idx = tid + 256 * ii;
      int c = idx >> 7, g = idx & 127;
      float tv = tbuf[c * FDIM + g] + tbuf[1024 + c * FDIM + g];
      float pr = tv * allb[(1 + c) * FDIM + g];
      if (c < 3) pP += pr; else dP += pr;
    }
    red[tid] = pP; red[256 + tid] = dP;
  }
  __syncthreads();
  for (int s = 128; s > 0; s >>= 1) {
    if (tid < s) { red[tid] += red[tid + s]; red[256 + tid] += red[256 + tid + s]; }
    __syncthreads();
  }

  if (tid < 128) {
    float cfea = ws[WS_FEAT + (size_t)i * FDIM + tid];   // table 0 = c_fea
    out[(size_t)i * FDIM + tid] = cfea + allb[tid] + red[0] + red[256];
  }
}

// ---------------- Launch ----------------
extern "C" void kernel_launch(void* const* d_in, const int* in_sizes, int n_in,
                              void* d_out, int out_size, void* d_ws, size_t ws_size,
                              hipStream_t stream) {
  const float* x    = (const float*)d_in[0];
  const int*   nbr  = (const int*)d_in[1];
  const float* gs   = (const float*)d_in[2];
  const float* gp   = (const float*)d_in[3];
  const float* gd   = (const float*)d_in[4];
  const float* Gs   = (const float*)d_in[5];
  const float* Gp   = (const float*)d_in[6];
  const float* Gd   = (const float*)d_in[7];
  const float* P1   = (const float*)d_in[8];
  const float* P2   = (const float*)d_in[9];
  const float* D1   = (const float*)d_in[10];
  const float* D2   = (const float*)d_in[11];
  const float* bng  = (const float*)d_in[12];
  const float* bnb  = (const float*)d_in[13];
  const float* resW = (const float*)d_in[14];
  const float* resb = (const float*)d_in[15];
  const float* linW = (const float*)d_in[16];
  const float* linb = (const float*)d_in[17];
  float* ws  = (float*)d_ws;
  float* out = (float*)d_out;

  bn_partial<<<BN_BLOCKS, 256, 0, stream>>>(x, ws);
  bn_finalize<<<1, 128, 0, stream>>>(bng, bnb, ws);
  qmat<<<256, 128, 0, stream>>>(P1, P2, D1, D2, ws);
  dim3 g4((NATOM + 127) / 128, 4);
  resmlp_wmma<<<g4, 256, 0, stream>>>(x, resW, resb, linW, linb, ws);
  env_final<<<NATOM, 256, 0, stream>>>(nbr, gs, gp, gd, Gs, Gp, Gd, ws, out);
}